// SE3Head_49770081026147
// MI455X (gfx1250) — compile-verified
//
#include <hip/hip_runtime.h>

typedef __attribute__((ext_vector_type(2))) float v2f;
typedef __attribute__((ext_vector_type(8))) float v8f;

#define NN1   1024                 // N+1
#define NN    1023                 // N
#define HH    32                   // heads
#define TJ    256                  // j-chunk staged in LDS
#define LPAD  34                   // LDS row stride in floats (even -> 8B-aligned b64 loads)
#define BLK   256
#define NWAVE 8
#define PLANE (NN1 * NN1)          // elements per head plane

// Issue one chunk's staging: this thread asynchronously copies column j = c0+tid
// (clamped) of all 32 head planes straight into LDS (transposed to [j][k]).
// 32 x global_load_async_to_lds_b32 per wave; 128B coalesced per instruction.
__device__ __forceinline__ void issue_chunk(const float* __restrict__ base,
                                            int c0, int tid, uint32_t la)
{
    // order this wave's prior ds_loads of the target buffer before overwrite
    asm volatile("s_wait_dscnt 0x0" ::: "memory");
    const int rem = NN - c0;
    const int ce  = (tid < rem) ? tid : 0;        // clamp padded lane (data discarded)
    const uint32_t voff = (uint32_t)(c0 + ce) * 4u;
#pragma unroll 4
    for (int h = 0; h < HH; ++h) {
        asm volatile(
            "global_load_async_to_lds_b32 %0, %1, %2 th:TH_LOAD_NT"
            :: "v"(la + (uint32_t)h * 4u),                       // LDS dest (bytes)
               "v"(voff + (uint32_t)h * (uint32_t)PLANE * 4u),   // global byte offset
               "s"(base)                                         // 64-bit SGPR base
            : "memory");
    }
}

__global__ __launch_bounds__(BLK)
void se3head_kernel(const float* __restrict__ x,
                    const float* __restrict__ qk,
                    const float* __restrict__ W1,
                    const float* __restrict__ b1,
                    const float* __restrict__ W2,
                    const float* __restrict__ b2p,
                    float* __restrict__ out)
{
    __shared__ float lds_q[2][TJ * LPAD];       // double-buffered q tiles: [j][k]
    __shared__ float lds_red[NWAVE * 4];        // cross-wave reduction

    const int tid  = threadIdx.x;
    const int lane = tid & 31;
    const int wave = tid >> 5;
    const int lh   = lane & 15;   // column index (g) / j-within-tile selector
    const int hi   = lane >> 4;   // half: selects K pair (A/B frags) and row set (D)

    const int i = blockIdx.x;     // 0..NN-1
    const int b = blockIdx.y;     // 0..1

    // ---- preload W1 B-fragments: B[k][g] = W1[g][k]; 8 k-chunks x 2 g-blocks ----
    v2f Bf[8][2];
#pragma unroll
    for (int kc = 0; kc < 8; ++kc) {
#pragma unroll
        for (int nb = 0; nb < 2; ++nb) {
            const int g  = 16 * nb + lh;
            const int k0 = 4 * kc + 2 * hi;
            Bf[kc][nb].x = W1[g * HH + k0];
            Bf[kc][nb].y = W1[g * HH + k0 + 1];
        }
    }
    const float b1v0 = b1[lh], b1v1 = b1[lh + 16];
    const float w2v0 = W2[lh], w2v1 = W2[lh + 16];
    const float b2   = b2p[0];

    // qk element (h=0, j) for this (b, i); head plane h adds h*PLANE
    const float* __restrict__ base =
        qk + (size_t)b * HH * PLANE + (size_t)(i + 1) * NN1 + 1;

    // LDS byte address of this thread's staging column in each buffer
    // (generic->LDS: low 32 bits of the flat address are the LDS offset)
    const uint32_t lds_col0 = (uint32_t)(uintptr_t)&lds_q[0][tid * LPAD];
    const uint32_t lds_col1 = (uint32_t)(uintptr_t)&lds_q[1][tid * LPAD];

    float S = 0.f, V0 = 0.f, V1 = 0.f, V2 = 0.f;

    // ---- wave-local software pipeline: no workgroup barriers in the main loop ----
    issue_chunk(base, 0, tid, lds_col0);

    for (int c0 = 0, nb = 0; c0 < NN; c0 += TJ, nb ^= 1) {
        const int next = c0 + TJ;
        if (next < NN) {
            issue_chunk(base, next, tid, nb ? lds_col0 : lds_col1);
            asm volatile("s_wait_asynccnt 0x20" ::: "memory");  // chunk c0 landed
        } else {
            asm volatile("s_wait_asynccnt 0x0" ::: "memory");
        }

        const float* __restrict__ qbuf = lds_q[nb];
#pragma unroll
        for (int t = 0; t < 2; ++t) {
            const int j0 = wave * 32 + t * 16;   // this wave's own staged columns
            const float* __restrict__ arow = &qbuf[(j0 + lh) * LPAD];

            v8f D0 = {0.f,0.f,0.f,0.f,0.f,0.f,0.f,0.f};
            v8f D1 = {0.f,0.f,0.f,0.f,0.f,0.f,0.f,0.f};
#pragma unroll
            for (int kc = 0; kc < 8; ++kc) {
                const int k0 = 4 * kc + 2 * hi;
                v2f A = *(const v2f*)(arow + k0);           // ds_load_b64
                if ((__float_as_uint(A.x) & 0x7fffffffu) == 0x7f800000u) A.x = 0.f;
                if ((__float_as_uint(A.y) & 0x7fffffffu) == 0x7f800000u) A.y = 0.f;
                D0 = __builtin_amdgcn_wmma_f32_16x16x4_f32(
                        false, A, false, Bf[kc][0], (short)0, D0, false, false);
                D1 = __builtin_amdgcn_wmma_f32_16x16x4_f32(
                        false, A, false, Bf[kc][1], (short)0, D1, false, false);
            }

            // bias + relu + W2 dot, folded per lane (this lane's g and g+16 columns)
            float p[8];
#pragma unroll
            for (int r = 0; r < 8; ++r) {
                float h0 = D0[r] + b1v0; h0 = h0 > 0.f ? h0 : 0.f;
                float h1 = D1[r] + b1v1; h1 = h1 > 0.f ? h1 : 0.f;
                p[r] = h0 * w2v0 + h1 * w2v1;
            }
            // reduce over g across the 16 lanes of each half (xor masks stay in-half)
#pragma unroll
            for (int r = 0; r < 8; ++r) {
                float s = p[r];
                s += __shfl_xor(s, 1, 32);
                s += __shfl_xor(s, 2, 32);
                s += __shfl_xor(s, 4, 32);
                s += __shfl_xor(s, 8, 32);
                p[r] = s;
            }

            // lanes with lh<8 fold row M = 8*hi + (lane&7) into S / V
            if (lh < 8) {
                const int rsel = lane & 7;
                float cv = p[0];
                cv = (rsel == 1) ? p[1] : cv;
                cv = (rsel == 2) ? p[2] : cv;
                cv = (rsel == 3) ? p[3] : cv;
                cv = (rsel == 4) ? p[4] : cv;
                cv = (rsel == 5) ? p[5] : cv;
                cv = (rsel == 6) ? p[6] : cv;
                cv = (rsel == 7) ? p[7] : cv;
                const int j = c0 + j0 + 8 * hi + rsel;
                if (j < NN) {
                    const float cj = cv + b2;
                    const float* __restrict__ xp = x + ((size_t)b * NN + j) * 3;
                    S  += cj;
                    V0 += cj * xp[0];
                    V1 += cj * xp[1];
                    V2 += cj * xp[2];
                }
            }
        }
    }

    // ---- wave reduction (inactive lanes hold 0) ----
#pragma unroll
    for (int m = 1; m <= 16; m <<= 1) {
        S  += __shfl_xor(S,  m, 32);
        V0 += __shfl_xor(V0, m, 32);
        V1 += __shfl_xor(V1, m, 32);
        V2 += __shfl_xor(V2, m, 32);
    }
    if (lane == 0) {
        lds_red[wave * 4 + 0] = S;
        lds_red[wave * 4 + 1] = V0;
        lds_red[wave * 4 + 2] = V1;
        lds_red[wave * 4 + 3] = V2;
    }
    __syncthreads();
    if (tid == 0) {
        float s = 0.f, v0 = 0.f, v1 = 0.f, v2 = 0.f;
#pragma unroll
        for (int w = 0; w < NWAVE; ++w) {
            s  += lds_red[w * 4 + 0];
            v0 += lds_red[w * 4 + 1];
            v1 += lds_red[w * 4 + 2];
            v2 += lds_red[w * 4 + 3];
        }
        const float* __restrict__ xi = x   + ((size_t)b * NN + i) * 3;
        float*       __restrict__ o  = out + ((size_t)b * NN + i) * 3;
        o[0] = xi[0] * (1.f + s) - v0;
        o[1] = xi[1] * (1.f + s) - v1;
        o[2] = xi[2] * (1.f + s) - v2;
    }
}

extern "C" void kernel_launch(void* const* d_in, const int* in_sizes, int n_in,
                              void* d_out, int out_size, void* d_ws, size_t ws_size,
                              hipStream_t stream) {
    const float* x   = (const float*)d_in[0];
    const float* qk  = (const float*)d_in[1];
    const float* W1  = (const float*)d_in[2];
    const float* b1  = (const float*)d_in[3];
    const float* W2  = (const float*)d_in[4];
    const float* b2  = (const float*)d_in[5];
    float* out = (float*)d_out;

    dim3 grid(NN, 2);
    dim3 block(BLK);
    se3head_kernel<<<grid, block, 0, stream>>>(x, qk, W1, b1, W2, b2, out);
}